// SparseNet_4200478015513
// MI455X (gfx1250) — compile-verified
//
#include <hip/hip_runtime.h>

// ---------------------------------------------------------------------------
// ISTA sparse coding on gfx1250 (MI455X), fp32 WMMA (V_WMMA_F32_16X16X4_F32).
// All four reduction-heavy ops run on the matrix pipe; VALU only does the
// 16-tap conv gather, prox/soft-threshold and deterministic norm reductions.
// ---------------------------------------------------------------------------

typedef __attribute__((ext_vector_type(2))) float v2f;
typedef __attribute__((ext_vector_type(8))) float v8f;

#define BATCH 64
#define NCH   64                 // K sparse channels
#define MDIM  32
#define NPIX  1024               // 32*32 positions per (b,k)
#define FDIM  8
#define PATCH 64                 // 8*8 filter taps
#define OUTW  70
#define NP    4900               // 70*70
#define NPLD  4912               // padded per-batch stride (multiple of 16)
#define M2    1024               // M*M (pred width)
#define R_N   (BATCH * NCH * NPIX)   // 4,194,304
#define T_N   (BATCH * PATCH * NPIX) // 4,194,304
#define UT_N  (NP * M2)              // 5,017,600
#define NBLK_UPD 4096
#define STEP  0.2f               // 2 * R_LR (factor 2 from d/dpred of sum-sq)
#define LMDA  0.005f
#define MAX_ITERS 50

// workspace layout (float offsets), total ~56.5 MB
#define OFF_UT   0
#define OFF_R    (OFF_UT + UT_N)
#define OFF_T    (OFF_R + R_N)
#define OFF_CT   (OFF_T + T_N)
#define OFF_GCT  (OFF_CT + BATCH * NPLD)
#define OFF_DIFF (OFF_GCT + BATCH * NPLD)
#define OFF_WT   (OFF_DIFF + BATCH * M2)
#define OFF_PART (OFF_WT + PATCH * NCH)
#define OFF_FLAG (OFF_PART + 2 * NBLK_UPD)

__device__ __forceinline__ v8f wmma4(v2f a, v2f b, v8f c) {
  // D = A(16x4,f32) * B(4x16,f32) + C(16x16,f32); CDNA5 matrix pipe.
  return __builtin_amdgcn_wmma_f32_16x16x4_f32(false, a, false, b,
                                               (short)0, c, false, false);
}

// WGP-scope (scope 0) prefetch: fills all cache levels on a miss.
// SE/DEV/SYS-scope prefetch only fills GL2, useless for L2-resident streams.
// WGP-scope is non-speculative, so callers keep addresses in-bounds.
__device__ __forceinline__ void prefetch_wgp(const void* p) {
  asm volatile("global_prefetch_b8 %0, off" : : "v"(p) : "memory");
}

// --- prep: U_T = U_w^T; wT = conv_w^T; zero R; clear converged flag ----------
__global__ void k_prep(const float* __restrict__ Uw, const float* __restrict__ w,
                       float* __restrict__ ws) {
  size_t idx = (size_t)blockIdx.x * blockDim.x + threadIdx.x;
  if (idx < (size_t)UT_N) {
    int m = (int)(idx & 1023);
    size_t c = idx >> 10;
    ws[OFF_UT + idx] = Uw[(size_t)m * NP + c];   // coalesced write
  }
  if (idx < (size_t)R_N) ws[OFF_R + idx] = 0.f;  // R0 = 0
  if (idx < (size_t)(PATCH * NCH)) {             // wT[f][k] = w[k][f]
    int f = (int)(idx >> 6), k = (int)(idx & 63);
    ws[OFF_WT + idx] = w[k * PATCH + f];
  }
  if (idx == 0) ((int*)(ws + OFF_FLAG))[0] = 0;
}

// --- stage 1a: T[b][f][p] = sum_k wT[f][k] * R[b][k][p]  (WMMA fp32) ---------
// Per image: M=64 taps, N=1024 positions, K=64 channels.
template <bool CHECK>
__global__ void k_tmat(float* __restrict__ ws) {
  if (CHECK && *(const volatile int*)(ws + OFF_FLAG)) return;
  int lane = threadIdx.x & 31;
  int wv = blockIdx.x * 4 + (threadIdx.x >> 5);        // 0..16383
  int ptile = wv & 63, ftile = (wv >> 6) & 3, b = wv >> 8;
  int h = lane >> 4, l15 = lane & 15;
  const float* Rb = ws + OFF_R + (size_t)b * NCH * NPIX;
  const float* arow = ws + OFF_WT + (ftile * 16 + l15) * NCH + 2 * h;
  int p = ptile * 16 + l15;
  v8f c = {0.f, 0.f, 0.f, 0.f, 0.f, 0.f, 0.f, 0.f};
  for (int k0 = 0; k0 < NCH; k0 += 4) {                // 16 WMMA steps
    v2f a = *(const v2f*)(arow + k0);                  // wT row, contiguous
    v2f bf;
    bf[0] = Rb[(size_t)(k0 + 2 * h + 0) * NPIX + p];   // coalesced across lanes
    bf[1] = Rb[(size_t)(k0 + 2 * h + 1) * NPIX + p];
    c = wmma4(a, bf, c);
  }
  float* Tb = ws + OFF_T + (size_t)b * PATCH * NPIX;
  for (int v = 0; v < 8; ++v)
    Tb[(size_t)(ftile * 16 + 8 * h + v) * NPIX + p] = c[v];
}

// --- stage 1b: ct[b,oy,ox] = bias + sum over <=16 valid taps of T ------------
template <bool CHECK>
__global__ void k_gather(const float* __restrict__ cb, float* __restrict__ ws) {
  if (CHECK && *(const volatile int*)(ws + OFF_FLAG)) return;
  int tid = blockIdx.x * blockDim.x + threadIdx.x;     // exactly BATCH*NP
  int b = tid / NP, pix = tid % NP;
  int oy = pix / OUTW, ox = pix % OUTW;
  int iy_lo = max(0, (oy - 6) / 2), iy_hi = min(MDIM - 1, oy >> 1);
  int ix_lo = max(0, (ox - 6) / 2), ix_hi = min(MDIM - 1, ox >> 1);
  const float* Tb = ws + OFF_T + (size_t)b * PATCH * NPIX;
  float acc = cb[0];
  for (int iy = iy_lo; iy <= iy_hi; ++iy) {
    int fy = oy - 2 * iy;
    for (int ix = ix_lo; ix <= ix_hi; ++ix) {
      int f = fy * FDIM + (ox - 2 * ix);
      acc += Tb[(size_t)f * NPIX + iy * MDIM + ix];
    }
  }
  ws[OFF_CT + (size_t)b * NPLD + pix] = acc;
}

// --- stage 2: pred = ct @ U_wT (M=64 batch, N=1024, K=4900), WMMA fp32 -------
// FINAL=false: write diff = pred - img into ws.  FINAL=true: write pred to out.
template <bool FINAL>
__global__ void k_pred(const float* __restrict__ img, float* __restrict__ out,
                       float* __restrict__ ws) {
  if (!FINAL && *(const volatile int*)(ws + OFF_FLAG)) return;
  int lane = threadIdx.x & 31;
  int wv = blockIdx.x * (blockDim.x >> 5) + (threadIdx.x >> 5);  // 0..255
  int btile = wv >> 6, ntile = wv & 63;
  int h = lane >> 4, l15 = lane & 15;
  const float* UT = ws + OFF_UT;
  int n = ntile * 16 + l15;
  const float* arow = ws + OFF_CT + (size_t)(btile * 16 + l15) * NPLD + 2 * h;
  v8f c = {0.f, 0.f, 0.f, 0.f, 0.f, 0.f, 0.f, 0.f};
  for (int c0 = 0; c0 < NP; c0 += 4) {                 // 1225 WMMA steps
    v2f a = *(const v2f*)(arow + c0);                  // A: lane row, K=2h..2h+1
    int kb = c0 + 2 * h;
    v2f bf;
    bf[0] = UT[(size_t)(kb + 0) * M2 + n];             // B: coalesced
    bf[1] = UT[(size_t)(kb + 1) * M2 + n];
    int pf = kb + 40; if (pf > NP - 2) pf = NP - 2;    // stay in-bounds
    prefetch_wgp(&UT[(size_t)(pf + 0) * M2 + n]);      // both B rows: 4KB apart
    prefetch_wgp(&UT[(size_t)(pf + 1) * M2 + n]);
    c = wmma4(a, bf, c);
  }
  for (int v = 0; v < 8; ++v) {
    int rb = btile * 16 + 8 * h + v;                   // batch row
    size_t o = (size_t)rb * M2 + n;
    if (FINAL) out[o] = c[v];
    else       ws[OFF_DIFF + o] = c[v] - img[o];
  }
}

// --- stage 3: g_ct = diff @ U_w (M=64, N=4900, K=1024), WMMA fp32 ------------
// Interior tiles (306/307) take a guard-free path; only the edge tile masks.
__global__ void k_gct(const float* __restrict__ Uw, float* __restrict__ ws) {
  if (*(const volatile int*)(ws + OFF_FLAG)) return;
  int lane = threadIdx.x & 31;
  int wv = blockIdx.x * (blockDim.x >> 5) + (threadIdx.x >> 5);  // 0..1227
  int btile = wv / 307, ntile = wv % 307;
  int h = lane >> 4, l15 = lane & 15;
  const float* arow = ws + OFF_DIFF + (size_t)(btile * 16 + l15) * M2 + 2 * h;
  int n = ntile * 16 + l15;
  v8f c = {0.f, 0.f, 0.f, 0.f, 0.f, 0.f, 0.f, 0.f};
  if (ntile != 306) {                                  // wave-uniform split
    const float* brow = Uw + (size_t)(2 * h) * NP + n;
    for (int k0 = 0; k0 < M2; k0 += 4) {               // 256 clean WMMA steps
      v2f a = *(const v2f*)(arow + k0);
      v2f bf;
      bf[0] = brow[(size_t)(k0 + 0) * NP];
      bf[1] = brow[(size_t)(k0 + 1) * NP];
      int pf = k0 + 40; if (pf > M2 - 44) pf = M2 - 44; // in-bounds (rows<1024)
      prefetch_wgp(&brow[(size_t)(pf + 0) * NP]);
      prefetch_wgp(&brow[(size_t)(pf + 1) * NP]);
      c = wmma4(a, bf, c);
    }
  } else {
    for (int k0 = 0; k0 < M2; k0 += 4) {
      v2f a = *(const v2f*)(arow + k0);
      v2f bf; bf[0] = 0.f; bf[1] = 0.f;
      if (n < NP) {                                    // mask N-edge (4900)
        bf[0] = Uw[(size_t)(k0 + 2 * h + 0) * NP + n];
        bf[1] = Uw[(size_t)(k0 + 2 * h + 1) * NP + n];
      }
      c = wmma4(a, bf, c);
    }
  }
  for (int v = 0; v < 8; ++v) {
    int rb = btile * 16 + 8 * h + v;
    ws[OFF_GCT + (size_t)rb * NPLD + n] = c[v];        // padded store, no guard
  }
}

// --- stage 4: grad_R = conv(g_ct, w, stride 2) as per-image GEMM + prox ------
// Per image b: C[k(16) x p(16)] tiles, K-dim = 64 filter taps (no boundary!).
__global__ void k_update(const float* __restrict__ w, float* __restrict__ ws) {
  if (*(const volatile int*)(ws + OFF_FLAG)) return;
  int lane = threadIdx.x & 31;
  int wv = blockIdx.x * 4 + (threadIdx.x >> 5);        // 0..16383
  int ptile = wv & 63, ktile = (wv >> 6) & 3, b = wv >> 8;
  int h = lane >> 4, l15 = lane & 15;
  const float* gct = ws + OFF_GCT + (size_t)b * NPLD;
  int p = ptile * 16 + l15, iy = p >> 5, ix = p & 31;
  const float* g0 = gct + (2 * iy) * OUTW + 2 * ix;
  const float* arow = w + (ktile * 16 + l15) * PATCH + 2 * h;
  v8f c = {0.f, 0.f, 0.f, 0.f, 0.f, 0.f, 0.f, 0.f};
  for (int f0 = 0; f0 < PATCH; f0 += 4) {              // 16 WMMA steps
    v2f a = *(const v2f*)(arow + f0);                  // A = w[k][f], float2
    int f = f0 + 2 * h;
    v2f bf;
    bf[0] = g0[(f >> 3) * OUTW + (f & 7)];             // coalesced (p per lane)
    bf[1] = g0[((f + 1) >> 3) * OUTW + ((f + 1) & 7)];
    c = wmma4(a, bf, c);
  }
  // prox (soft threshold) + in-place R update + norm partials
  float snum = 0.f, sden = 0.f;
  float* Rp = ws + OFF_R + (size_t)b * NCH * NPIX;
  for (int v = 0; v < 8; ++v) {
    int k = ktile * 16 + 8 * h + v;
    size_t o = (size_t)k * NPIX + p;
    float r  = Rp[o];
    float x  = r - STEP * c[v];
    float ax = fabsf(x) - LMDA;
    float rn = ax > 0.f ? copysignf(ax, x) : 0.f;
    float d  = rn - r;
    snum += d * d;  sden += r * r;
    Rp[o] = rn;
  }
  __shared__ float sn[128], sd[128];
  sn[threadIdx.x] = snum; sd[threadIdx.x] = sden;
  __syncthreads();
  for (int s = 64; s > 0; s >>= 1) {
    if (threadIdx.x < s) {
      sn[threadIdx.x] += sn[threadIdx.x + s];
      sd[threadIdx.x] += sd[threadIdx.x + s];
    }
    __syncthreads();
  }
  if (threadIdx.x == 0) {
    ws[OFF_PART + 2 * blockIdx.x + 0] = sn[0];
    ws[OFF_PART + 2 * blockIdx.x + 1] = sd[0];
  }
}

// --- stage 5: deterministic fixed-order reduction + convergence flag ---------
__global__ void k_reduce(float* __restrict__ ws) {
  int* flag = (int*)(ws + OFF_FLAG);
  if (*(volatile int*)flag) return;
  __shared__ float sn[256], sd[256];
  float a = 0.f, b = 0.f;
  for (int i = threadIdx.x; i < NBLK_UPD; i += 256) {  // fixed per-thread order
    a += ws[OFF_PART + 2 * i + 0];
    b += ws[OFF_PART + 2 * i + 1];
  }
  sn[threadIdx.x] = a; sd[threadIdx.x] = b;
  __syncthreads();
  for (int s = 128; s > 0; s >>= 1) {
    if (threadIdx.x < s) {
      sn[threadIdx.x] += sn[threadIdx.x + s];
      sd[threadIdx.x] += sd[threadIdx.x + s];
    }
    __syncthreads();
  }
  // ratio < 0.01  <=>  num < 1e-4 * den  (den==0 -> inf -> keep going; NaN ok)
  if (threadIdx.x == 0 && sn[0] < 1e-4f * sd[0]) *flag = 1;
}

extern "C" void kernel_launch(void* const* d_in, const int* in_sizes, int n_in,
                              void* d_out, int out_size, void* d_ws, size_t ws_size,
                              hipStream_t stream) {
  const float* img = (const float*)d_in[0];   // (64, 1024)
  const float* Uw  = (const float*)d_in[1];   // (1024, 4900)
  const float* w   = (const float*)d_in[2];   // (64, 1, 8, 8)
  const float* cb  = (const float*)d_in[3];   // (1,)
  float* ws  = (float*)d_ws;                  // needs ~57 MB
  float* out = (float*)d_out;                 // (64, 1024)

  k_prep<<<(UT_N + 255) / 256, 256, 0, stream>>>(Uw, w, ws);

  for (int it = 0; it < MAX_ITERS; ++it) {
    k_tmat<true><<<NBLK_UPD, 128, 0, stream>>>(ws);
    k_gather<true><<<(BATCH * NP) / 256, 256, 0, stream>>>(cb, ws);
    k_pred<false><<<64, 128, 0, stream>>>(img, out, ws);
    k_gct<<<307, 128, 0, stream>>>(Uw, ws);
    k_update<<<NBLK_UPD, 128, 0, stream>>>(w, ws);
    k_reduce<<<1, 256, 0, stream>>>(ws);
  }

  // final predict(R_final) -> d_out (unguarded)
  k_tmat<false><<<NBLK_UPD, 128, 0, stream>>>(ws);
  k_gather<false><<<(BATCH * NP) / 256, 256, 0, stream>>>(cb, ws);
  k_pred<true><<<64, 128, 0, stream>>>(img, out, ws);
}